// uSNC_conv_53575422050638
// MI455X (gfx1250) — compile-verified
//
#include <hip/hip_runtime.h>
#include <cstdint>

#define EPSF 1e-20f
#define HH 768
#define WW 768
#define BB 4
#define HWSZ (HH * WW)
#define TILE 32

// ---- CDNA5 async global->LDS path (gfx1250) -------------------------------
// Generic LDS pointers on gfx1250 are {SHARED_BASE[63:32], lds_offset[31:0]},
// so truncating to 32 bits yields the wave-relative LDS byte address the
// async-load instruction wants in VDST.
__device__ __forceinline__ void async_ld_f32(const float* g, void* l) {
    uint32_t loff = (uint32_t)(uintptr_t)l;
    asm volatile("global_load_async_to_lds_b32 %0, %1, off"
                 :: "v"(loff), "v"(g)
                 : "memory");
}
__device__ __forceinline__ void async_wait0() {
    asm volatile("s_wait_asynccnt 0x0" ::: "memory");
}

// ---- kernel 0: weight preparation (tiny, one wave) ------------------------
__global__ void kprep(const float* __restrict__ wcd_in, const float* __restrict__ wsd_in,
                      const float* __restrict__ wpow_in, const float* __restrict__ wprop_in,
                      const float* __restrict__ wcs_in, const float* __restrict__ wss_in,
                      float* __restrict__ wb) {
    if (threadIdx.x != 0) return;
    auto sp = [](float v) { return log1pf(expf(v)); };
    // d-side spatial weights: (3,2) softplus, mirror col0 -> (3,3), normalize
    float wd[9]; float sum = 0.f;
    for (int h = 0; h < 3; ++h) {
        float a = sp(wsd_in[h * 2 + 0]);
        float b = sp(wsd_in[h * 2 + 1]);
        wd[h * 3 + 0] = a; wd[h * 3 + 1] = b; wd[h * 3 + 2] = a;
    }
    for (int k = 0; k < 9; ++k) sum += wd[k];
    float wcd = sp(wcd_in[0]);
    float wcdn = wcd / wcd;               // (1,1) channel weight normalizes to 1
    for (int k = 0; k < 9; ++k) wb[k] = wcdn * wd[k] / sum;
    // s-side
    float ws9[9]; sum = 0.f;
    for (int h = 0; h < 3; ++h) {
        float a = sp(wss_in[h * 2 + 0]);
        float b = sp(wss_in[h * 2 + 1]);
        ws9[h * 3 + 0] = a; ws9[h * 3 + 1] = b; ws9[h * 3 + 2] = a;
    }
    for (int k = 0; k < 9; ++k) sum += ws9[k];
    float wcs = sp(wcs_in[0]);
    float wcsn = wcs / wcs;
    for (int k = 0; k < 9; ++k) wb[9 + k] = wcsn * ws9[k] / sum;
    wb[18] = 1.f / (1.f + expf(-wprop_in[0]));   // wps (sigmoid)
    wb[19] = sp(wpow_in[0]);                     // wpow (softplus)
}

// ---- kernel 1: argmax/argmin + blend + 3x3 conv -> scs_out, cs_out --------
__global__ __launch_bounds__(256) void k1(const float* __restrict__ x,
                                          const float* __restrict__ scs,
                                          const float* __restrict__ cs,
                                          const float* __restrict__ wb,
                                          float* __restrict__ out) {
    const int b = blockIdx.z;
    const int ty0 = blockIdx.y * TILE, tx0 = blockIdx.x * TILE;
    const int tid = threadIdx.x;
    __shared__ float sD[36 * 37];   // dcd, later d = dcd/(cd+eps), halo-2 region
    __shared__ float sC[36 * 37];   // cd
    __shared__ float sMc[34 * 35];  // cs_mid, halo-1 region
    __shared__ float sMs[34 * 35];  // scs_mid
    const float* dcd = x + (size_t)b * HWSZ;
    const float* cd  = x + (size_t)(BB + b) * HWSZ;

    // async-stage dcd & cd 36x36 halo regions into LDS (zero-fill OOB)
    for (int n = tid; n < 36 * 36 * 2; n += 256) {
        int plane = n / (36 * 36);
        int rem = n % (36 * 36);
        int r = rem / 36, c = rem % 36;
        int gy = ty0 - 2 + r, gx = tx0 - 2 + c;
        float* dst = (plane ? sC : sD) + r * 37 + c;
        if ((unsigned)gy < HH && (unsigned)gx < WW) {
            const float* src = (plane ? cd : dcd) + gy * WW + gx;
            async_ld_f32(src, dst);
        } else {
            *dst = 0.f;
        }
    }
    async_wait0();
    __syncthreads();

    // convert sD in place to the ratio d = dcd/(cd+eps) (OOB stays 0/(0+eps)=0)
    for (int n = tid; n < 36 * 36; n += 256) {
        int i = (n / 36) * 37 + (n % 36);
        sD[i] = sD[i] / (sC[i] + EPSF);
    }
    __syncthreads();

    const float wps  = wb[18];
    const float wpow = wb[19];
    const float omw  = 1.f - wps;

    // stage A: mids over the 34x34 halo-1 region
    for (int n = tid; n < 34 * 34; n += 256) {
        int qr = n / 34, qc = n % 34;
        float vmax = -1.f, vmin = -1.f;
        float djmax = 0.f, cjmax = 0.f, djmin = 0.f, cjmin = 0.f;
        #pragma unroll
        for (int i = 0; i < 3; ++i) {
            #pragma unroll
            for (int j = 0; j < 3; ++j) {
                float d = sD[(qr + i) * 37 + (qc + j)];
                float c = sC[(qr + i) * 37 + (qc + j)];
                float pm = d * c;
                if (pm > vmax) { vmax = pm; djmax = d; cjmax = c; }
                float pr = c / (d + EPSF);
                if (pr > vmin) { vmin = pr; djmin = d; cjmin = c; }
            }
        }
        float s_new  = powf((djmin + EPSF) / (djmax + EPSF), wpow);
        float cs_new = cjmax * cjmin;
        int gy = ty0 - 1 + qr, gx = tx0 - 1 + qc;
        float mc = 0.f, ms = 0.f;
        if ((unsigned)gy < HH && (unsigned)gx < WW) {
            size_t gi = (size_t)b * HWSZ + (size_t)gy * WW + gx;
            mc = cs[gi]  * wps + cs_new * omw;
            ms = scs[gi] * wps + s_new * cs_new * omw;
        }
        sMc[qr * 35 + qc] = mc;
        sMs[qr * 35 + qc] = ms;
    }
    __syncthreads();

    // stage B: 3x3 conv of mids -> scs_out, cs_out (both are final outputs)
    float wk[9];
    #pragma unroll
    for (int k = 0; k < 9; ++k) wk[k] = wb[9 + k];
    for (int n = tid; n < TILE * TILE; n += 256) {
        int pr = n / TILE, pc = n % TILE;
        float co = 0.f, so = 0.f;
        #pragma unroll
        for (int i = 0; i < 3; ++i) {
            #pragma unroll
            for (int j = 0; j < 3; ++j) {
                float w = wk[i * 3 + j];
                co = fmaf(w, sMc[(pr + i) * 35 + (pc + j)], co);
                so = fmaf(w, sMs[(pr + i) * 35 + (pc + j)], so);
            }
        }
        size_t gi = (size_t)b * HWSZ + (size_t)(ty0 + pr) * WW + (tx0 + pc);
        out[(size_t)8 * HWSZ + gi]  = so;   // scs_out
        out[(size_t)12 * HWSZ + gi] = co;   // cs_out
    }
}

// ---- kernel 2: s = scs_out/(cs_out+eps), per-pixel weights, 9-tap sums ----
__global__ __launch_bounds__(256) void k2(const float* __restrict__ x,
                                          const float* __restrict__ wb,
                                          float* __restrict__ out) {
    const int b = blockIdx.z;
    const int ty0 = blockIdx.y * TILE, tx0 = blockIdx.x * TILE;
    const int tid = threadIdx.x;
    __shared__ float sS[34 * 35];   // scs_out, later s = scs_out/(cs_out+eps)
    __shared__ float sT[34 * 35];   // cs_out
    __shared__ float sD[34 * 35];   // raw dcd
    __shared__ float sC[34 * 35];   // raw cd
    const float* dcd  = x + (size_t)b * HWSZ;
    const float* cd   = x + (size_t)(BB + b) * HWSZ;
    const float* scsO = out + (size_t)8 * HWSZ  + (size_t)b * HWSZ;  // written by k1
    const float* csO  = out + (size_t)12 * HWSZ + (size_t)b * HWSZ;  // written by k1

    for (int n = tid; n < 34 * 34 * 4; n += 256) {
        int plane = n / (34 * 34);
        int rem = n % (34 * 34);
        int r = rem / 34, c = rem % 34;
        float* dst = (plane == 0 ? sS : plane == 1 ? sT : plane == 2 ? sD : sC) + r * 35 + c;
        int gy = ty0 - 1 + r, gx = tx0 - 1 + c;
        if ((unsigned)gy < HH && (unsigned)gx < WW) {
            const float* src =
                (plane == 0 ? scsO : plane == 1 ? csO : plane == 2 ? dcd : cd) + gy * WW + gx;
            async_ld_f32(src, dst);
        } else {
            *dst = 0.f;
        }
    }
    async_wait0();
    __syncthreads();

    // convert sS in place to s = scs_out/(cs_out+eps)  (OOB stays 0 -> pad1(s))
    for (int n = tid; n < 34 * 34; n += 256) {
        int i = (n / 34) * 35 + (n % 34);
        sS[i] = sS[i] / (sT[i] + EPSF);
    }
    __syncthreads();

    float wk[9];
    #pragma unroll
    for (int k = 0; k < 9; ++k) wk[k] = wb[k];
    for (int n = tid; n < TILE * TILE; n += 256) {
        int pr = n / TILE, pc = n % TILE;
        float scen = sS[(pr + 1) * 35 + (pc + 1)];
        float num_d = 0.f, num_c = 0.f, den = EPSF;
        #pragma unroll
        for (int i = 0; i < 3; ++i) {
            #pragma unroll
            for (int j = 0; j < 3; ++j) {
                int k = i * 3 + j;
                int li = (pr + i) * 35 + (pc + j);
                float wsv = (k == 4) ? 1.0f : sS[li] * scen;
                float t = wsv * wk[k];
                den += t;
                num_d = fmaf(t, sD[li], num_d);
                num_c = fmaf(t, sC[li], num_c);
            }
        }
        float inv = 1.f / den;
        size_t gi = (size_t)(ty0 + pr) * WW + (tx0 + pc);
        out[((size_t)2 * b) * HWSZ + gi]     = num_d * inv;  // x_out interleaved: 2b
        out[((size_t)2 * b + 1) * HWSZ + gi] = num_c * inv;  // 2b+1
    }
}

extern "C" void kernel_launch(void* const* d_in, const int* in_sizes, int n_in,
                              void* d_out, int out_size, void* d_ws, size_t ws_size,
                              hipStream_t stream) {
    (void)in_sizes; (void)n_in; (void)out_size; (void)ws_size;
    const float* x            = (const float*)d_in[0];
    const float* scs          = (const float*)d_in[1];
    const float* cs           = (const float*)d_in[2];
    const float* w_channel_d  = (const float*)d_in[3];
    const float* w_spatial_d  = (const float*)d_in[4];
    const float* w_pow_s      = (const float*)d_in[5];
    const float* w_prop_s     = (const float*)d_in[6];
    const float* w_channel_s  = (const float*)d_in[7];
    const float* w_spatial_s  = (const float*)d_in[8];
    float* out = (float*)d_out;
    float* wb  = (float*)d_ws;                 // 20 prepared weight scalars

    kprep<<<1, 32, 0, stream>>>(w_channel_d, w_spatial_d, w_pow_s, w_prop_s,
                                w_channel_s, w_spatial_s, wb);
    dim3 grid(WW / TILE, HH / TILE, BB);
    k1<<<grid, 256, 0, stream>>>(x, scs, cs, wb, out);
    k2<<<grid, 256, 0, stream>>>(x, wb, out);
}